// DotProductAttention_317827580734
// MI455X (gfx1250) — compile-verified
//
#include <hip/hip_runtime.h>
#include <hip/hip_bf16.h>

typedef __attribute__((ext_vector_type(2))) float v2f;
typedef __attribute__((ext_vector_type(4))) float v4f;
typedef __attribute__((ext_vector_type(8))) float v8f;

#define BATCH     32
#define QLEN      2048
#define KLEN      2048
#define DDIM      64
#define QT        16          // query rows per block
#define SE_STRIDE 2052        // padded LDS row stride (floats): 4*m bank skew -> conflict-free b64 reads
#define SQ_STRIDE 68
#define SC_STRIDE 68
#define NWAVE     8
#define NTHREADS  256

// LDS layout (floats):
//   sQ    : QT * SQ_STRIDE                    = 1088
//   sE    : QT * SE_STRIDE                    = 32832
//   sP    : NWAVE * QT * SC_STRIDE            = 8704   (per-wave GEMM2 partials)
//   sPart : QT * 16                           = 256    (row-sum partials)
//   sInv  : QT                                = 16
#define LDS_FLOATS (QT*SQ_STRIDE + QT*SE_STRIDE + NWAVE*QT*SC_STRIDE + QT*16 + QT)

__global__ __launch_bounds__(NTHREADS)
void attn_fused_kernel(const float* __restrict__ Qm,
                       const float* __restrict__ Km,
                       const float* __restrict__ Vm,
                       float* __restrict__ ctx_out,
                       float* __restrict__ attn_out)
{
    extern __shared__ float smem[];
    float* sQ    = smem;
    float* sE    = sQ + QT * SQ_STRIDE;
    float* sP    = sE + QT * SE_STRIDE;
    float* sPart = sP + NWAVE * QT * SC_STRIDE;
    float* sInv  = sPart + QT * 16;

    const int tid  = threadIdx.x;
    const int lane = tid & 31;
    const int wv   = tid >> 5;
    const int m    = lane & 15;     // A-frag row / B-frag col
    const int half = lane >> 4;     // 0 or 1
    const int kkb  = half * 2;      // K sub-offset within 4-wide step

    const int bq = blockIdx.x;
    const int b  = bq >> 7;                 // 128 q-tiles per batch
    const int q0 = (bq & 127) * QT;

    const float* Qp = Qm + ((size_t)b * QLEN + q0) * DDIM;
    const float* Kp = Km + (size_t)b * KLEN * DDIM;
    const float* Vp = Vm + (size_t)b * KLEN * DDIM;

    // ---- Stage 0: load Q tile (16x64) into LDS ----
    {
        int r  = tid >> 4;
        int d4 = (tid & 15) * 4;
        v4f q = *(const v4f*)(Qp + r * DDIM + d4);
        *(v4f*)(sQ + r * SQ_STRIDE + d4) = q;
    }
    __syncthreads();

    // ---- Stage 1: GEMM1  S = Q * K^T, then E = exp(S) -> sE ----
    v2f aQ[16];
#pragma unroll
    for (int ks = 0; ks < 16; ++ks)
        aQ[ks] = *(const v2f*)(sQ + m * SQ_STRIDE + ks * 4 + kkb);

    for (int t = 0; t < 16; ++t) {
        int n0 = wv * 256 + t * 16;                 // key-column base of this 16x16 tile
        v8f acc = {};
        const float* kb = Kp + (size_t)(n0 + m) * DDIM + kkb;
#pragma unroll
        for (int ks = 0; ks < 16; ++ks) {
            v2f bf = *(const v2f*)(kb + ks * 4);
            acc = __builtin_amdgcn_wmma_f32_16x16x4_f32(false, aQ[ks], false, bf,
                                                        (short)0, acc, false, false);
        }
#pragma unroll
        for (int j = 0; j < 8; ++j) {
            int row = half * 8 + j;
            sE[row * SE_STRIDE + n0 + m] = __expf(acc[j]);  // scores |s|<~46: safe in f32
        }
    }
    __syncthreads();

    // ---- Stage 2: deterministic row sums -> sInv ----
    {
        int row = tid >> 4;
        int cl  = tid & 15;
        float part = 0.0f;
        for (int i = 0; i < 32; ++i) {
            v4f v = *(const v4f*)(sE + row * SE_STRIDE + cl * 4 + i * 64);
            part += v.x + v.y + v.z + v.w;
        }
        sPart[row * 16 + cl] = part;
    }
    __syncthreads();
    if (tid < QT) {
        float s = 0.0f;
#pragma unroll
        for (int i = 0; i < 16; ++i) s += sPart[tid * 16 + i];
        sInv[tid] = 1.0f / s;
    }
    __syncthreads();

    // ---- Stage 3: stream normalized attention (the 512MB write) ----
    // Non-temporal stores: this stream is write-once and must not evict the
    // L2-resident K/V working set (32MB) out of the 192MB global L2.
    {
        float* pA = attn_out + ((size_t)b * QLEN + q0) * KLEN;
        for (int i = 0; i < 32; ++i) {
            int idx = tid + i * NTHREADS;            // over [16 rows][512 float4 chunks]
            int row = idx >> 9;
            int ch  = idx & 511;
            v4f v = *(const v4f*)(sE + row * SE_STRIDE + ch * 4);
            float s = sInv[row];
            v4f o = { v.x * s, v.y * s, v.z * s, v.w * s };
            __builtin_nontemporal_store(o, (v4f*)(pA + (size_t)row * KLEN + ch * 4));
        }
    }

    // ---- Stage 4: GEMM2  Ctx = E * V, K split across 8 waves ----
    {
        int kbase = wv * 256;
        float* myP = sP + wv * QT * SC_STRIDE;
        for (int ct = 0; ct < 4; ++ct) {
            v8f acc = {};
            const float* vb = Vp + (size_t)(kbase + kkb) * DDIM + ct * 16 + m;
#pragma unroll 4
            for (int ks = 0; ks < 64; ++ks) {
                v2f af = *(const v2f*)(sE + m * SE_STRIDE + kbase + ks * 4 + kkb);
                v2f bf;
                bf.x = vb[(size_t)(ks * 4)     * DDIM];
                bf.y = vb[(size_t)(ks * 4 + 1) * DDIM];
                acc = __builtin_amdgcn_wmma_f32_16x16x4_f32(false, af, false, bf,
                                                            (short)0, acc, false, false);
            }
#pragma unroll
            for (int j = 0; j < 8; ++j) {
                int row = half * 8 + j;
                myP[row * SC_STRIDE + ct * 16 + m] = acc[j];   // unique slot: deterministic
            }
        }
    }
    __syncthreads();

    // ---- Stage 5: deterministic 8-way reduce, scale by 1/rowsum, store context ----
    for (int idx = tid; idx < QT * DDIM; idx += NTHREADS) {
        int row = idx >> 6;
        int d   = idx & 63;
        float s = 0.0f;
#pragma unroll
        for (int w = 0; w < NWAVE; ++w)
            s += sP[(w * QT + row) * SC_STRIDE + d];
        __builtin_nontemporal_store(s * sInv[row],
            ctx_out + ((size_t)b * QLEN + q0 + row) * DDIM + d);
    }
}

extern "C" void kernel_launch(void* const* d_in, const int* in_sizes, int n_in,
                              void* d_out, int out_size, void* d_ws, size_t ws_size,
                              hipStream_t stream) {
    (void)in_sizes; (void)n_in; (void)out_size; (void)d_ws; (void)ws_size;
    const float* Qm = (const float*)d_in[0];
    const float* Km = (const float*)d_in[1];
    const float* Vm = (const float*)d_in[2];
    float* ctx  = (float*)d_out;                                  // [B,Q,D] first
    float* attn = ctx + (size_t)BATCH * QLEN * DDIM;              // then [B,Q,K]

    size_t shmem = (size_t)LDS_FLOATS * sizeof(float);            // ~172 KB (<=320KB/WGP)
    hipFuncSetAttribute((const void*)attn_fused_kernel,
                        hipFuncAttributeMaxDynamicSharedMemorySize, (int)shmem);

    dim3 grid(BATCH * (QLEN / QT));                               // 4096 blocks
    attn_fused_kernel<<<grid, NTHREADS, shmem, stream>>>(Qm, Km, Vm, ctx, attn);
}